// DifferentiableMatcher_249108103937
// MI455X (gfx1250) — compile-verified
//
#include <hip/hip_runtime.h>
#include <hip/hip_bf16.h>
#include <math.h>

typedef __attribute__((ext_vector_type(2))) float v2f;
typedef __attribute__((ext_vector_type(8))) float v8f;

#define KDIM   (512 * 512)   // reduction length (H*W)
#define MROWS  100           // pred masks
#define NROWS  50            // gt masks
#define MTILES 7             // ceil(100/16)
#define KSPLIT 512           // workgroups along K
#define KC     (KDIM / KSPLIT) // 512 K elements per workgroup

// ---------------------------------------------------------------- zero accum
__global__ void dm_zero(float* __restrict__ acc) {
    int i = blockIdx.x * blockDim.x + threadIdx.x;
    if (i < MROWS * NROWS) acc[i] = 0.0f;
}

// ---------------------------------------------------------------- K-split GEMM
// acc[m][n] += sum_{k in chunk} pred[m][k] * gt[n][k]
__global__ __launch_bounds__(128) void dm_gemm(const float* __restrict__ pred,
                                               const float* __restrict__ gt,
                                               float* __restrict__ acc) {
    const int lane  = threadIdx.x & 31;
    const int wave  = threadIdx.x >> 5;   // 0..3 -> N tile
    const int n0    = wave * 16;
    const int lhalf = lane >> 4;          // 0 or 1 (selects K sub-pair)
    const int l15   = lane & 15;

    const int kbase = blockIdx.x * KC;
    // Both A and B fragments read a float2 at column kbase + k + 2*lhalf
    const int colbase = kbase + 2 * lhalf;

    // B fragment source row (gt row), clamped into valid range
    int nrow = n0 + l15; if (nrow > NROWS - 1) nrow = NROWS - 1;
    const float* __restrict__ gtrow = gt + (size_t)nrow * KDIM + colbase;

    // A fragment source rows per M tile, clamped
    const float* prow[MTILES];
#pragma unroll
    for (int mt = 0; mt < MTILES; ++mt) {
        int m = mt * 16 + l15; if (m > MROWS - 1) m = MROWS - 1;
        prow[mt] = pred + (size_t)m * KDIM + colbase;
    }

    v8f c[MTILES] = {};  // 7 x 16x16 f32 accumulators

    // Unroll 8: one base address per row reused across 8 K-steps via
    // IOFFSET immediates (offset:0..112), minimizing 64-bit address VALU.
#pragma unroll 8
    for (int k = 0; k < KC; k += 4) {
        v2f b = *(const v2f*)(gtrow + k);
#pragma unroll
        for (int mt = 0; mt < MTILES; ++mt) {
            v2f a = *(const v2f*)(prow[mt] + k);
            c[mt] = __builtin_amdgcn_wmma_f32_16x16x4_f32(
                /*neg_a=*/false, a, /*neg_b=*/false, b,
                /*c_mod=*/(short)0, c[mt], /*reuse_a=*/false, /*reuse_b=*/false);
        }
    }

    // C/D layout: VGPR r -> (M = r + 8*lhalf, N = l15) within the 16x16 tile
    const int n = n0 + l15;
#pragma unroll
    for (int mt = 0; mt < MTILES; ++mt) {
#pragma unroll
        for (int r = 0; r < 8; ++r) {
            int m = mt * 16 + r + 8 * lhalf;
            if (m < MROWS && n < NROWS) {
                __hip_atomic_fetch_add(&acc[m * NROWS + n], c[mt][r],
                                       __ATOMIC_RELAXED, __HIP_MEMORY_SCOPE_AGENT);
            }
        }
    }
}

// ---------------------------------------------------------------- Sinkhorn + exp
__global__ __launch_bounds__(128) void dm_sinkhorn(const float* __restrict__ acc,
                                                   float* __restrict__ out) {
    __shared__ float la[MROWS * NROWS];
    const int tid = threadIdx.x;

    // log_alpha = -cost/TEMP = (dot - 1) * 10
    for (int i = tid; i < MROWS * NROWS; i += 128)
        la[i] = (acc[i] - 1.0f) * 10.0f;
    __syncthreads();

    for (int it = 0; it < 5; ++it) {
        // logsumexp over n (axis=2), one thread per row
        if (tid < MROWS) {
            float* row = la + tid * NROWS;
            float mx = -INFINITY;
            for (int n = 0; n < NROWS; ++n) mx = fmaxf(mx, row[n]);
            float s = 0.0f;
            for (int n = 0; n < NROWS; ++n) s += __expf(row[n] - mx);
            float lse = mx + __logf(s);
            for (int n = 0; n < NROWS; ++n) row[n] -= lse;
        }
        __syncthreads();
        // logsumexp over k (axis=1), one thread per column
        if (tid < NROWS) {
            float mx = -INFINITY;
            for (int m = 0; m < MROWS; ++m) mx = fmaxf(mx, la[m * NROWS + tid]);
            float s = 0.0f;
            for (int m = 0; m < MROWS; ++m) s += __expf(la[m * NROWS + tid] - mx);
            float lse = mx + __logf(s);
            for (int m = 0; m < MROWS; ++m) la[m * NROWS + tid] -= lse;
        }
        __syncthreads();
    }

    for (int i = tid; i < MROWS * NROWS; i += 128)
        out[i] = __expf(la[i]);
}

// ---------------------------------------------------------------- launch
extern "C" void kernel_launch(void* const* d_in, const int* in_sizes, int n_in,
                              void* d_out, int out_size, void* d_ws, size_t ws_size,
                              hipStream_t stream) {
    const float* pred = (const float*)d_in[0]; // [1,100,512,512]
    const float* gt   = (const float*)d_in[1]; // [50,512,512]
    float* out = (float*)d_out;                // [1,100,50]
    float* acc = (float*)d_ws;                 // 100*50 f32 accumulator

    dm_zero<<<(MROWS * NROWS + 255) / 256, 256, 0, stream>>>(acc);
    dm_gemm<<<KSPLIT, 128, 0, stream>>>(pred, gt, acc);
    dm_sinkhorn<<<1, 128, 0, stream>>>(acc, out);
}